// PaPsLoss_22539988370116
// MI455X (gfx1250) — compile-verified
//
#include <hip/hip_runtime.h>
#include <hip/hip_bf16.h>
#include <math.h>

// Problem constants (match the reference)
#define NB_B 8
#define NH   256
#define NW   256
#define NS   64
#define NCLS_ 20
#define NK   2048
#define EPSF 1e-8f

#define NPIX (NB_B * NH * NW)   // 524288
#define NBC       128           // blocks for center-loss pass
#define CENTER_T  256
#define KT        128           // threads per per-k block

typedef __attribute__((ext_vector_type(2))) float v2f;
typedef __attribute__((ext_vector_type(8))) float v8f;

#if defined(__HIP_DEVICE_COMPILE__) && __has_builtin(__builtin_amdgcn_wmma_f32_16x16x4_f32)
#define USE_WMMA_REDUCE 1
#else
#define USE_WMMA_REDUCE 0
#endif

// Wave-level sum of one f32 per lane.
// WMMA path: D = ones(16x4) x B(4x16) => every D element is a column sum of B.
// The 64-element B operand (2 VGPRs x 32 lanes) holds {v, 0} per lane, so the
// sum of the 16 column sums equals the sum of all 32 lane values. Executed by
// every wave with EXEC all-1s (WMMA requirement).
__device__ __forceinline__ float wave_sum(float v) {
#if USE_WMMA_REDUCE
  v2f a; a[0] = 1.0f; a[1] = 1.0f;
  v2f b; b[0] = v;    b[1] = 0.0f;
  v8f c = {};
  c = __builtin_amdgcn_wmma_f32_16x16x4_f32(false, a, false, b, (short)0, c,
                                            false, false);
  float s = c[0];                 // lane L (0..15): column sum N=L (M=0 row)
  s += __shfl_xor(s, 1, 32);
  s += __shfl_xor(s, 2, 32);
  s += __shfl_xor(s, 4, 32);
  s += __shfl_xor(s, 8, 32);      // lanes 0..15 now hold the full wave total
  return s;
#else
  float s = v;
  #pragma unroll
  for (int m = 16; m >= 1; m >>= 1) s += __shfl_xor(s, m, 32);
  return s;
#endif
}

// Block-level sum; `sh` must hold >= blockDim.x/32 floats.
__device__ __forceinline__ float block_sum(float v, float* sh) {
  const int t  = threadIdx.x;
  const int nw = blockDim.x >> 5;
  float w = wave_sum(v);
  if ((t & 31) == 0) sh[t >> 5] = w;
  __syncthreads();
  float tot;
  if (t == 0) {
    tot = 0.0f;
    for (int i = 0; i < nw; ++i) tot += sh[i];
    sh[0] = tot;
  }
  __syncthreads();
  tot = sh[0];
  __syncthreads();                // allow safe reuse of sh
  return tot;
}

// ---------------- Kernel 1: center loss partials (+ optional plane extract) --
// Channel 0 (gt heatmap) and channel 1 (instance id) are 4 bytes apart, so the
// cacheline is already resident: extracting the dense channel-1 plane here is
// free bandwidth-wise and turns the shape pass's 28B-stride gathers into
// dense 4B-stride reads (~7x less L2 traffic on the dominant pass).
template <bool WRITE_PLANE>
__global__ void paps_center_kernel(const float* __restrict__ hm,
                                   const float* __restrict__ tgt,
                                   float* __restrict__ cpart,
                                   float* __restrict__ plane) {
  __shared__ float sh[CENTER_T / 32];
  float pos = 0.0f, neg = 0.0f, cnt = 0.0f;
  for (int i = blockIdx.x * blockDim.x + threadIdx.x; i < NPIX;
       i += gridDim.x * blockDim.x) {
    float p = hm[i];                       // (B,1,H,W) flat == (B,H,W) flat
    float g = tgt[(size_t)i * 7 + 0];      // channel 0
    if (WRITE_PLANE) plane[i] = tgt[(size_t)i * 7 + 1];
    if (g == 1.0f) {
      pos += logf(p + EPSF);
      cnt += 1.0f;
    } else if (g < 1.0f) {
      float s1 = 1.0f - g;
      float s2 = s1 * s1;
      neg += logf(1.0f - p + EPSF) * (s2 * s2);   // (1-g)^4, BETA=4
    }
  }
  pos = block_sum(pos, sh);
  neg = block_sum(neg, sh);
  cnt = block_sum(cnt, sh);
  if (threadIdx.x == 0) {
    cpart[blockIdx.x * 3 + 0] = pos;
    cpart[blockIdx.x * 3 + 1] = neg;
    cpart[blockIdx.x * 3 + 2] = cnt;
  }
}

// ---------------- Kernel 2: per-k size / class / shape partials ----------------
template <bool USE_PLANE>
__global__ void paps_perk_kernel(const float* __restrict__ size_pred,
                                 const float* __restrict__ sem,
                                 const float* __restrict__ masks,
                                 const float* __restrict__ tgt,
                                 const float* __restrict__ plane,
                                 const int*  __restrict__ cbij,
                                 const int*  __restrict__ boxes,
                                 float* __restrict__ kpart) {
  __shared__ float sh[KT / 32];
  const int k   = blockIdx.x;
  const int b   = cbij[k * 3 + 0];
  const int ci  = cbij[k * 3 + 1];
  const int cj  = cbij[k * 3 + 2];
  const int xtl = boxes[k * 4 + 0];   // boxes[:,0] = jj (x)
  const int ytl = boxes[k * 4 + 1];   // boxes[:,1] = ii (y)
  const size_t cidx = ((size_t)(b * NH + ci) * NW + cj) * 7;
  const float zid = tgt[cidx + 2];

  // shape: sum over 64x64 of lse(m,1-m) - (crop==zid ? m : 1-m)
  float ssum = 0.0f;
  const float* mk = masks + (size_t)k * (NS * NS);
  for (int i = threadIdx.x; i < NS * NS; i += blockDim.x) {
    const int r  = i >> 6;
    const int cl = i & 63;
    float m  = mk[i];
    const size_t pix = (size_t)(b * NH + (ytl + r)) * NW + (xtl + cl);
    float tv = USE_PLANE ? plane[pix] : tgt[pix * 7 + 1];
    float om = 1.0f - m;
    float hi = fmaxf(m, om), lo = fminf(m, om);
    float lse = hi + log1pf(expf(lo - hi));       // log(e^m + e^(1-m))
    float sel = (tv == zid) ? m : om;
    ssum += lse - sel;
  }
  ssum = block_sum(ssum, sh);

  if (threadIdx.x == 0) {
    // size term: sum_c |ts - sp| / (ts + eps)
    float ts0 = tgt[cidx + 3], ts1 = tgt[cidx + 4];
    float sp0 = size_pred[k * 2 + 0], sp1 = size_pred[k * 2 + 1];
    float sterm = fabsf(ts0 - sp0) / (ts0 + EPSF) +
                  fabsf(ts1 - sp1) / (ts1 + EPSF);
    // class term: -log_softmax(logits)[label]  (gamma = 0)
    int lbl = (int)tgt[cidx + 5];
    const float* lg = sem + (size_t)k * NCLS_;
    float mx = lg[0];
    #pragma unroll
    for (int c = 1; c < NCLS_; ++c) mx = fmaxf(mx, lg[c]);
    float se = 0.0f;
    #pragma unroll
    for (int c = 0; c < NCLS_; ++c) se += expf(lg[c] - mx);
    float logpt = lg[lbl] - mx - logf(se);

    kpart[k * 3 + 0] = sterm;
    kpart[k * 3 + 1] = -logpt;
    kpart[k * 3 + 2] = ssum;
  }
}

// ---------------- Kernel 3: final combine ----------------
__global__ void paps_final_kernel(const float* __restrict__ cpart,
                                  const float* __restrict__ kpart,
                                  float* __restrict__ out) {
  __shared__ float sh[256 / 32];
  const int t = threadIdx.x;
  float pos = 0.f, neg = 0.f, cnt = 0.f;
  float ssum = 0.f, csum = 0.f, shsum = 0.f;
  for (int i = t; i < NBC; i += 256) {
    pos += cpart[i * 3 + 0];
    neg += cpart[i * 3 + 1];
    cnt += cpart[i * 3 + 2];
  }
  for (int i = t; i < NK; i += 256) {
    ssum  += kpart[i * 3 + 0];
    csum  += kpart[i * 3 + 1];
    shsum += kpart[i * 3 + 2];
  }
  pos   = block_sum(pos, sh);
  neg   = block_sum(neg, sh);
  cnt   = block_sum(cnt, sh);
  ssum  = block_sum(ssum, sh);
  csum  = block_sum(csum, sh);
  shsum = block_sum(shsum, sh);
  if (t == 0) {
    float loss_center = -(pos + neg) / cnt;
    float loss_size   = ssum / (float)NK;
    float loss_class  = csum / (float)NK;
    float loss_shape  = shsum / ((float)NK * (float)(NS * NS));
    out[0] = loss_center + loss_size + loss_class + loss_shape;
  }
}

extern "C" void kernel_launch(void* const* d_in, const int* in_sizes, int n_in,
                              void* d_out, int out_size, void* d_ws,
                              size_t ws_size, hipStream_t stream) {
  (void)in_sizes; (void)n_in; (void)out_size;
  const float* heatmap   = (const float*)d_in[0];
  const float* size_pred = (const float*)d_in[1];
  const float* sem       = (const float*)d_in[2];
  const float* masks     = (const float*)d_in[3];
  const float* tgt       = (const float*)d_in[4];
  const int*   cbij      = (const int*)d_in[5];
  const int*   boxes     = (const int*)d_in[6];
  float* out = (float*)d_out;

  // Workspace layout. Plane path needs (NPIX + NBC*3 + NK*3) floats (~2.1 MB);
  // fall back to strided gathers if scratch is smaller. ws_size is fixed per
  // run, so this branch is deterministic.
  const size_t need_plane =
      ((size_t)NPIX + (size_t)NBC * 3 + (size_t)NK * 3) * sizeof(float);
  const bool use_plane = (ws_size >= need_plane);

  float* plane = (float*)d_ws;                          // NPIX floats (if used)
  float* cpart = use_plane ? plane + NPIX : (float*)d_ws;  // NBC*3 floats
  float* kpart = cpart + NBC * 3;                          // NK*3 floats

  if (use_plane) {
    paps_center_kernel<true><<<NBC, CENTER_T, 0, stream>>>(heatmap, tgt, cpart,
                                                           plane);
    paps_perk_kernel<true><<<NK, KT, 0, stream>>>(size_pred, sem, masks, tgt,
                                                  plane, cbij, boxes, kpart);
  } else {
    paps_center_kernel<false><<<NBC, CENTER_T, 0, stream>>>(heatmap, tgt,
                                                            cpart, nullptr);
    paps_perk_kernel<false><<<NK, KT, 0, stream>>>(size_pred, sem, masks, tgt,
                                                   nullptr, cbij, boxes, kpart);
  }
  paps_final_kernel<<<1, 256, 0, stream>>>(cpart, kpart, out);
}